// Resizing1D_11201274708407
// MI455X (gfx1250) — compile-verified
//
#include <hip/hip_runtime.h>
#include <stdint.h>

// Problem constants (from the reference)
#define TL      1000      // TARGET_LEN
#define S_LEN   4096
#define CH      128
#define BATCH   64
#define WPB     8         // waves per block (256 threads, wave32)
#define RPW     4         // output rows pipelined per wave
#define ROWS_TOTAL (BATCH * TL)                    // 64000
#define GRID_X     (ROWS_TOTAL / (WPB * RPW))      // 2000 (exact)

#if defined(__HIP_DEVICE_COMPILE__) && defined(__gfx1250__) && \
    __has_builtin(__builtin_amdgcn_global_load_async_to_lds_b128)
#define USE_ASYNC 1
#else
#define USE_ASYNC 0
#endif

#if USE_ASYNC
typedef int v4i __attribute__((ext_vector_type(4)));
typedef __attribute__((address_space(1))) v4i* gptr_v4i;
typedef __attribute__((address_space(3))) v4i* lptr_v4i;
#endif

// ---------------------------------------------------------------------------
// Kernel 1: lengths[b] = (last index where mask[b,:] is true) + 1, or S if none.
// One block per sample; each thread owns exactly one 16-byte chunk (256*16=4096).
// ---------------------------------------------------------------------------
__global__ __launch_bounds__(256) void resz_len_kernel(
    const unsigned char* __restrict__ mask, int* __restrict__ lengths) {
  __shared__ int smax;
  const int b = blockIdx.x;
  if (threadIdx.x == 0) smax = -1;
  __syncthreads();
  const uint4* row = (const uint4*)(mask + (size_t)b * S_LEN);
  const uint4 v = row[threadIdx.x];  // single coalesced b128 load
  const unsigned w[4] = {v.x, v.y, v.z, v.w};
  int local = -1;
#pragma unroll
  for (int k = 0; k < 4; ++k)
    if (w[k] != 0u)  // highest nonzero byte within the word (little-endian)
      local = (int)(threadIdx.x * 16 + k * 4 + (3 - (__builtin_clz(w[k]) >> 3)));
  if (local >= 0) atomicMax(&smax, local);
  __syncthreads();
  if (threadIdx.x == 0) lengths[b] = (smax < 0) ? S_LEN : (smax + 1);
}

// ---------------------------------------------------------------------------
// Kernel 2: gather+lerp resize. One wave per RPW (4) output rows; lane covers 4 channels
// (32 lanes x float4 = 512B = one channel row). Async-LDS double-buffered
// pipeline on gfx1250; direct float4 loads otherwise.
// ---------------------------------------------------------------------------
__global__ __launch_bounds__(256) void resz_resize_kernel(
    const float* __restrict__ in, const int* __restrict__ lengths,
    float* __restrict__ out) {
  const int wave = threadIdx.x >> 5;
  const int lane = threadIdx.x & 31;
  const int row0 = (blockIdx.x * WPB + wave) * RPW;

  // Per-row source coordinates, exactly as the reference computes them.
  auto coords = [&](int r, int& b, int& lo, int& hi, float& w) {
    b = r / TL;
    const int l = r - b * TL;
    const int len = lengths[b];
    const float scale = (float)len * (1.0f / (float)TL);
    const float src = ((float)l + 0.5f) * scale - 0.5f;
    int lof = (int)floorf(src);
    int hic = (int)ceilf(src);
    const int mx = len - 1;
    lo = lof < 0 ? 0 : (lof > mx ? mx : lof);
    hi = hic < 0 ? 0 : (hic > mx ? mx : hic);
    w = src - (float)lo;  // weight uses the CLAMPED lo (matches TF kernel)
  };

#if USE_ASYNC
  __shared__ float buf[WPB][2][2][CH];  // [wave][pingpong][lo/hi][channels]
  float wgt[RPW];

  auto issue = [&](int i) {
    int b, lo, hi;
    float w;
    coords(row0 + i, b, lo, hi, w);
    wgt[i] = w;
    const float* glo = in + ((size_t)b * S_LEN + lo) * CH + (lane << 2);
    const float* ghi = in + ((size_t)b * S_LEN + hi) * CH + (lane << 2);
    __builtin_amdgcn_global_load_async_to_lds_b128(
        (gptr_v4i)glo, (lptr_v4i)&buf[wave][i & 1][0][lane << 2], 0, 0);
    __builtin_amdgcn_global_load_async_to_lds_b128(
        (gptr_v4i)ghi, (lptr_v4i)&buf[wave][i & 1][1][lane << 2], 0, 0);
  };

  auto compute = [&](int i) {
    const int r = row0 + i;
    const float4 a = *(const float4*)&buf[wave][i & 1][0][lane << 2];
    const float4 c = *(const float4*)&buf[wave][i & 1][1][lane << 2];
    const float wi = wgt[i];
    float4 o;
    o.x = a.x + (c.x - a.x) * wi;
    o.y = a.y + (c.y - a.y) * wi;
    o.z = a.z + (c.z - a.z) * wi;
    o.w = a.w + (c.w - a.w) * wi;
    *(float4*)&out[(size_t)r * CH + (lane << 2)] = o;
  };

  issue(0);
#pragma unroll
  for (int i = 0; i < RPW; ++i) {
    if (i + 1 < RPW) {
      // WAR guard: issue(i+1) reuses the buffer compute(i-1) read from LDS.
      // The lerp's register deps mean these reads are long done -> ~free.
      if (i >= 1) asm volatile("s_wait_dscnt 0" ::: "memory");
      issue(i + 1);  // prefetch next row into the other buffer
      // wait until only the 2 just-issued loads remain outstanding
      asm volatile("s_wait_asynccnt 2" ::: "memory");
    } else {
      asm volatile("s_wait_asynccnt 0" ::: "memory");
    }
    compute(i);
  }
#else
  // Fallback: direct coalesced float4 gathers (host pass / missing builtin).
#pragma unroll
  for (int i = 0; i < RPW; ++i) {
    const int r = row0 + i;
    int b, lo, hi;
    float w;
    coords(r, b, lo, hi, w);
    const float4 a =
        *(const float4*)(in + ((size_t)b * S_LEN + lo) * CH + (lane << 2));
    const float4 c =
        *(const float4*)(in + ((size_t)b * S_LEN + hi) * CH + (lane << 2));
    float4 o;
    o.x = a.x + (c.x - a.x) * w;
    o.y = a.y + (c.y - a.y) * w;
    o.z = a.z + (c.z - a.z) * w;
    o.w = a.w + (c.w - a.w) * w;
    *(float4*)&out[(size_t)r * CH + (lane << 2)] = o;
  }
#endif
}

// ---------------------------------------------------------------------------
extern "C" void kernel_launch(void* const* d_in, const int* in_sizes, int n_in,
                              void* d_out, int out_size, void* d_ws,
                              size_t ws_size, hipStream_t stream) {
  const float* in = (const float*)d_in[0];
  const unsigned char* mask = (const unsigned char*)d_in[1];  // jax bool = u8
  int* lengths = (int*)d_ws;  // 64 ints of scratch
  float* out = (float*)d_out;

  resz_len_kernel<<<BATCH, 256, 0, stream>>>(mask, lengths);
  resz_resize_kernel<<<GRID_X, 256, 0, stream>>>(in, lengths, out);
}